// HeatTransferNetwork_73031623901252
// MI455X (gfx1250) — compile-verified
//
#include <hip/hip_runtime.h>
#include <cmath>

#define N_NODES 16384
#define NH_PTS  65536
#define E_EDGES 262144
#define HDIM    128
#define ZW      896          // 4*128 (Hk) + 128 (A) + 128 (B) + 128 (R)
#define ZW3     64           // layer-3 Z stride (7 live cols zero-padded to 64)
#define FLT_BIG 3.402823466e+38f

typedef __attribute__((ext_vector_type(2))) float v2f;
typedef __attribute__((ext_vector_type(8))) float v8f;

// ---------------- small utility kernels ----------------

__global__ void fill_kernel(float* __restrict__ p, int n, float v) {
    int i = blockIdx.x * blockDim.x + threadIdx.x;
    if (i < n) p[i] = v;
}

__global__ void prep_alphas(const float* __restrict__ a1, const float* __restrict__ a2,
                            const float* __restrict__ a3, float* __restrict__ alpha) {
    int l = threadIdx.x;
    if (l >= 3) return;
    const float* a = (l == 0) ? a1 : (l == 1) ? a2 : a3;
    float m = fmaxf(fmaxf(a[0], a[1]), fmaxf(a[2], a[3]));
    float e0 = __expf(a[0] - m), e1 = __expf(a[1] - m),
          e2 = __expf(a[2] - m), e3 = __expf(a[3] - m);
    float s = e0 + e1 + e2 + e3;
    alpha[l * 4 + 0] = e0 / s; alpha[l * 4 + 1] = e1 / s;
    alpha[l * 4 + 2] = e2 / s; alpha[l * 4 + 3] = e3 / s;
}

// logical weight value at (k, j) for layer 2's concatenated 128x896 matrix
__device__ __forceinline__ float wcat2_val(const float* __restrict__ Wk,
                                           const float* __restrict__ Wr,
                                           const float* __restrict__ We, int k, int j) {
    if (j < 512) { int m = j >> 7, c = j & 127; return Wk[m * 16384 + k * 128 + c]; }
    if (j < 640) { int c = j - 512; return We[k * 128 + c] - We[(128 + k) * 128 + c]; }
    if (j < 768) { int c = j - 640; return We[(128 + k) * 128 + c]; }
    int c = j - 768; return Wr[k * 128 + c];
}

// B stored directly in WMMA fragment order:
//   float index = ((tile*32 + s)*32 + lane)*2 + p,  k = 4s + 2*(lane>>4) + p,  col = tile*16 + (lane&15)
// -> GEMM loads one coalesced b64 per fragment at an immediate offset.
__global__ void build_wcat2_frag(const float* __restrict__ Wk, const float* __restrict__ Wr,
                                 const float* __restrict__ We, float* __restrict__ Wf) {
    int idx = blockIdx.x * 256 + threadIdx.x;          // 56*32*32*2 = 114688 threads
    int p = idx & 1, lane = (idx >> 1) & 31, s = (idx >> 6) & 31, t = idx >> 11;
    int k = 4 * s + ((lane >> 4) << 1) + p;
    int j = t * 16 + (lane & 15);
    Wf[idx] = wcat2_val(Wk, Wr, We, k, j);
}

// layer-3 weights, 128 x 64 zero-padded (cols: 0..3 Wk3[m][k], 4 Wet-Web, 5 Web, 6 Wr)
__global__ void build_wcat3_frag(const float* __restrict__ Wk, const float* __restrict__ Wr,
                                 const float* __restrict__ We, float* __restrict__ Wf) {
    int idx = blockIdx.x * 256 + threadIdx.x;          // 4*32*32*2 = 8192 threads
    int p = idx & 1, lane = (idx >> 1) & 31, s = (idx >> 6) & 31, t = idx >> 11;
    int k = 4 * s + ((lane >> 4) << 1) + p;
    int j = t * 16 + (lane & 15);
    float v = 0.0f;
    if (j < 4)       v = Wk[j * 128 + k];
    else if (j == 4) v = We[k] - We[128 + k];
    else if (j == 5) v = We[128 + k];
    else if (j == 6) v = Wr[k];
    Wf[idx] = v;
}

// Layer-1 node transform: ci==1, rank-1 -> no GEMM needed
__global__ void layer1_node(const float* __restrict__ x, const float* __restrict__ Wk,
                            const float* __restrict__ We, const float* __restrict__ Wr,
                            float* __restrict__ Z) {
    int i = blockIdx.x, j = threadIdx.x;
    float xv = x[i];
    float w;
    if (j < 512) { int m = j >> 7, c = j & 127; w = Wk[m * 128 + c]; }
    else if (j < 640) { int c = j - 512; w = We[c] - We[128 + c]; }
    else if (j < 768) { int c = j - 640; w = We[128 + c]; }
    else { int c = j - 768; w = Wr[c]; }
    Z[(size_t)i * ZW + j] = xv * w;
}

__global__ void deg_count(const int* __restrict__ ei, float* __restrict__ deg) {
    int e = blockIdx.x * blockDim.x + threadIdx.x;
    if (e < E_EDGES) atomicAdd(&deg[ei[E_EDGES + e]], 1.0f);
}

// ---------------- WMMA GEMM: Out(M x Ncols) = X(M x 128) @ W(128 x Ncols) ----------------
// One wave per 16x64 block (4 N-tiles): A fragment loaded once per k-step, fed to 4 WMMAs.
// B is pre-swizzled into fragment order -> 1 b64 load per fragment, immediate offsets only.
// Ncols must be a multiple of 64.
__global__ void gemm_wmma_f32_blocked(const float* __restrict__ X, const float* __restrict__ Wf,
                                      float* __restrict__ Out, int Ncols) {
    int lane = threadIdx.x & 31;
    int wv   = threadIdx.x >> 5;
    int ngroups = Ncols >> 6;
    int g = blockIdx.y * 4 + wv;
    if (g >= ngroups) return;                 // wave-uniform exit: EXEC all-ones for WMMA
    int tm = blockIdx.x;
    int mrow = (tm << 4) + (lane & 15);       // A: lane&15 = M row
    int koff = (lane >> 4) << 1;              // half-wave picks K pair
    const v2f* xr = (const v2f*)(X + (size_t)mrow * HDIM + koff);
    const v2f* wb = (const v2f*)Wf + (size_t)g * 4096 + lane;   // frag(t,s) at wb[t*1024 + s*32]
    v8f acc0 = {}, acc1 = {}, acc2 = {}, acc3 = {};
#pragma unroll
    for (int s = 0; s < 32; ++s) {
        v2f a  = xr[2 * s];                   // X[mrow][4s+koff .. +1], one b64, imm offset
        v2f b0 = wb[s * 32];
        v2f b1 = wb[1024 + s * 32];
        v2f b2 = wb[2048 + s * 32];
        v2f b3 = wb[3072 + s * 32];
        acc0 = __builtin_amdgcn_wmma_f32_16x16x4_f32(false, a, false, b0, (short)0, acc0, false, false);
        acc1 = __builtin_amdgcn_wmma_f32_16x16x4_f32(false, a, false, b1, (short)0, acc1, false, false);
        acc2 = __builtin_amdgcn_wmma_f32_16x16x4_f32(false, a, false, b2, (short)0, acc2, false, false);
        acc3 = __builtin_amdgcn_wmma_f32_16x16x4_f32(false, a, false, b3, (short)0, acc3, false, false);
    }
    int rbase = (lane >> 4) << 3;             // D layout: VGPR r -> M=r / M=r+8 per half-wave
    float* ob = Out + (size_t)(tm * 16 + rbase) * Ncols + g * 64 + (lane & 15);
#pragma unroll
    for (int r = 0; r < 8; ++r) {
        ob[(size_t)r * Ncols +  0] = acc0[r];
        ob[(size_t)r * Ncols + 16] = acc1[r];
        ob[(size_t)r * Ncols + 32] = acc2[r];
        ob[(size_t)r * Ncols + 48] = acc3[r];
    }
}

// ---------------- edge phase ----------------

__device__ __forceinline__ void atomicMaxF(float* addr, float v) {
    if (v >= 0.0f) atomicMax((int*)addr, __float_as_int(v));
    else           atomicMin((unsigned int*)addr, __float_as_uint(v));
}

// co=128: one wave per edge, each lane handles 4 channels (float4)
__global__ void edge128(const float* __restrict__ Z, const int* __restrict__ ei,
                        const float* __restrict__ ea, const float* __restrict__ alpha,
                        float* __restrict__ mk, float* __restrict__ em) {
    int lane = threadIdx.x & 31;
    int e = blockIdx.x * 8 + (threadIdx.x >> 5);
    int src = ei[e], dst = ei[E_EDGES + e];
    float d = ea[e];
    float a0 = alpha[0], a1 = alpha[1], a2 = alpha[2], a3 = alpha[3];
    float dd = d * d;
    float c1 = a1 * d, c2 = a2 * dd, c3 = a3 * dd * d;   // c0 = a0 (d^0 == 1)
    const float4* zs = (const float4*)(Z + (size_t)src * ZW);
    const float4* zd = (const float4*)(Z + (size_t)dst * ZW);
    float4 h0 = zs[lane], h1 = zs[32 + lane], h2 = zs[64 + lane], h3 = zs[96 + lane];
    float4 A  = zd[128 + lane];   // cols 512..639
    float4 B  = zs[160 + lane];   // cols 640..767
    float4 v;
    v.x = a0 * h0.x + c1 * h1.x + c2 * h2.x + c3 * h3.x;
    v.y = a0 * h0.y + c1 * h1.y + c2 * h2.y + c3 * h3.y;
    v.z = a0 * h0.z + c1 * h1.z + c2 * h2.z + c3 * h3.z;
    v.w = a0 * h0.w + c1 * h1.w + c2 * h2.w + c3 * h3.w;
    float* mkd = mk + (size_t)dst * HDIM + lane * 4;
    atomicAdd(mkd + 0, v.x); atomicAdd(mkd + 1, v.y);
    atomicAdd(mkd + 2, v.z); atomicAdd(mkd + 3, v.w);
    float* emd = em + (size_t)dst * HDIM + lane * 4;
    atomicMaxF(emd + 0, A.x + B.x); atomicMaxF(emd + 1, A.y + B.y);
    atomicMaxF(emd + 2, A.z + B.z); atomicMaxF(emd + 3, A.w + B.w);
}

// co=1 (layer 3): one thread per edge
__global__ void edge1(const float* __restrict__ Z, const int* __restrict__ ei,
                      const float* __restrict__ ea, const float* __restrict__ alpha,
                      float* __restrict__ mk, float* __restrict__ em) {
    int e = blockIdx.x * blockDim.x + threadIdx.x;
    if (e >= E_EDGES) return;
    int src = ei[e], dst = ei[E_EDGES + e];
    float d = ea[e];
    float dd = d * d;
    const float* zs = Z + (size_t)src * ZW3;
    float val = alpha[0] * zs[0] + alpha[1] * d * zs[1] + alpha[2] * dd * zs[2]
              + alpha[3] * dd * d * zs[3];
    atomicAdd(&mk[dst], val);
    atomicMaxF(&em[dst], Z[(size_t)dst * ZW3 + 4] + zs[5]);
}

// ---------------- combine ----------------

__global__ void combine128(const float* __restrict__ mk, const float* __restrict__ em,
                           const float* __restrict__ Z, const float* __restrict__ deg,
                           const float* __restrict__ b, int act, float* __restrict__ X) {
    int i = blockIdx.x, c = threadIdx.x;
    float degc = fmaxf(deg[i], 1.0f);
    float mv = mk[(size_t)i * HDIM + c] / degc;
    float ev = em[(size_t)i * HDIM + c];
    ev = (ev > -FLT_BIG) ? ev : 0.0f;          // segment_max of empty segment -> 0
    float o = mv + ev + Z[(size_t)i * ZW + 768 + c] + b[c];
    if (act) o = fmaxf(o, 0.0f);
    X[(size_t)i * HDIM + c] = o;
}

__global__ void combine1(const float* __restrict__ mk, const float* __restrict__ em,
                         const float* __restrict__ Z, const float* __restrict__ deg,
                         const float* __restrict__ b, float* __restrict__ e3) {
    int i = blockIdx.x * blockDim.x + threadIdx.x;
    if (i >= N_NODES) return;
    float degc = fmaxf(deg[i], 1.0f);
    float ev = em[i];
    ev = (ev > -FLT_BIG) ? ev : 0.0f;
    e3[i] = mk[i] / degc + ev + Z[(size_t)i * ZW3 + 6] + b[0];
}

// ---------------- fused kNN(k=4) + inverse-d2 interpolation of (x + e3) ----------------
// Both knn_interpolate calls share pos/pos_high -> identical idx/weights, computed once.
__global__ void knn_out_k(const float* __restrict__ ph, const float* __restrict__ pos,
                          const float* __restrict__ xin, const float* __restrict__ e3,
                          float* __restrict__ out) {
    int q = blockIdx.x * blockDim.x + threadIdx.x;
    float qx = ph[2 * q], qy = ph[2 * q + 1];
    const float2* p2 = (const float2*)pos;     // uniform index -> scalar (SMEM) loads
    float b0 = FLT_BIG, b1 = FLT_BIG, b2 = FLT_BIG, b3 = FLT_BIG;
    int i0 = 0, i1 = 0, i2 = 0, i3 = 0;
#pragma unroll 4
    for (int j = 0; j < N_NODES; ++j) {
        float2 p = p2[j];
        float dx = qx - p.x, dy = qy - p.y;
        float dd = fmaf(dx, dx, dy * dy);
        if (dd < b3) {                          // cheap reject: insertion path is rare
            if (dd < b2) {
                b3 = b2; i3 = i2;
                if (dd < b1) {
                    b2 = b1; i2 = i1;
                    if (dd < b0) { b1 = b0; i1 = i0; b0 = dd; i0 = j; }
                    else         { b1 = dd; i1 = j; }
                } else { b2 = dd; i2 = j; }
            } else { b3 = dd; i3 = j; }
        }
    }
    float w0 = 1.0f / (b0 + 1e-16f), w1 = 1.0f / (b1 + 1e-16f),
          w2 = 1.0f / (b2 + 1e-16f), w3 = 1.0f / (b3 + 1e-16f);
    float s = w0 * (xin[i0] + e3[i0]) + w1 * (xin[i1] + e3[i1])
            + w2 * (xin[i2] + e3[i2]) + w3 * (xin[i3] + e3[i3]);
    out[q] = s / (w0 + w1 + w2 + w3);
}

// ---------------- launch ----------------

// workspace layout (float offsets)
#define OFF_ALPHA 0
#define OFF_DEG   16
#define OFF_WCAT2 16400                          // 16 + 16384
#define OFF_WCAT3 131088                         // + 56*32*32*2 (=114688)
#define OFF_X     139280                         // + 4*32*32*2  (=8192)
#define OFF_Z     2236432                        // + 16384*128
#define OFF_MK    16916496                       // + 16384*896
#define OFF_EM    19013648                       // + 16384*128
#define OFF_MK3   21110800                       // + 16384*128
#define OFF_EM3   21127184
#define OFF_E3    21143568

extern "C" void kernel_launch(void* const* d_in, const int* in_sizes, int n_in,
                              void* d_out, int out_size, void* d_ws, size_t ws_size,
                              hipStream_t stream) {
    const float* x        = (const float*)d_in[0];
    const float* pos      = (const float*)d_in[1];
    const float* pos_high = (const float*)d_in[2];
    const float* eattr    = (const float*)d_in[3];
    const int*   ei       = (const int*)d_in[4];
    const float* Wk1 = (const float*)d_in[5],  *Wr1 = (const float*)d_in[6];
    const float* We1 = (const float*)d_in[7],  *b1  = (const float*)d_in[8];
    const float* al1 = (const float*)d_in[9];
    const float* Wk2 = (const float*)d_in[10], *Wr2 = (const float*)d_in[11];
    const float* We2 = (const float*)d_in[12], *b2  = (const float*)d_in[13];
    const float* al2 = (const float*)d_in[14];
    const float* Wk3 = (const float*)d_in[15], *Wr3 = (const float*)d_in[16];
    const float* We3 = (const float*)d_in[17], *b3  = (const float*)d_in[18];
    const float* al3 = (const float*)d_in[19];

    float* ws    = (float*)d_ws;
    float* alpha = ws + OFF_ALPHA;
    float* deg   = ws + OFF_DEG;
    float* Wcat2 = ws + OFF_WCAT2;
    float* Wcat3 = ws + OFF_WCAT3;
    float* X     = ws + OFF_X;
    float* Z     = ws + OFF_Z;
    float* MK    = ws + OFF_MK;
    float* EM    = ws + OFF_EM;
    float* MK3   = ws + OFF_MK3;
    float* EM3   = ws + OFF_EM3;
    float* E3v   = ws + OFF_E3;

    const float NEGINF = -INFINITY;

    prep_alphas<<<1, 32, 0, stream>>>(al1, al2, al3, alpha);
    build_wcat2_frag<<<448, 256, 0, stream>>>(Wk2, Wr2, We2, Wcat2);
    build_wcat3_frag<<<32, 256, 0, stream>>>(Wk3, Wr3, We3, Wcat3);

    fill_kernel<<<N_NODES / 256, 256, 0, stream>>>(deg, N_NODES, 0.0f);
    deg_count<<<E_EDGES / 256, 256, 0, stream>>>(ei, deg);

    // ---- layer 1 (ci=1 -> rank-1 node transform) ----
    layer1_node<<<N_NODES, ZW, 0, stream>>>(x, Wk1, We1, Wr1, Z);
    fill_kernel<<<(N_NODES * HDIM) / 256, 256, 0, stream>>>(MK, N_NODES * HDIM, 0.0f);
    fill_kernel<<<(N_NODES * HDIM) / 256, 256, 0, stream>>>(EM, N_NODES * HDIM, NEGINF);
    edge128<<<E_EDGES / 8, 256, 0, stream>>>(Z, ei, eattr, alpha + 0, MK, EM);
    combine128<<<N_NODES, HDIM, 0, stream>>>(MK, EM, Z, deg, b1, 1, X);

    // ---- layer 2 (WMMA GEMM 16384x128 @ 128x896, 14 groups of 64 cols) ----
    gemm_wmma_f32_blocked<<<dim3(N_NODES / 16, 4), 128, 0, stream>>>(X, Wcat2, Z, ZW);
    fill_kernel<<<(N_NODES * HDIM) / 256, 256, 0, stream>>>(MK, N_NODES * HDIM, 0.0f);
    fill_kernel<<<(N_NODES * HDIM) / 256, 256, 0, stream>>>(EM, N_NODES * HDIM, NEGINF);
    edge128<<<E_EDGES / 8, 256, 0, stream>>>(Z, ei, eattr, alpha + 4, MK, EM);
    combine128<<<N_NODES, HDIM, 0, stream>>>(MK, EM, Z, deg, b2, 1, X);

    // ---- layer 3 (WMMA GEMM 16384x128 @ 128x64, cols 7..63 zero-padded) ----
    gemm_wmma_f32_blocked<<<dim3(N_NODES / 16, 1), 128, 0, stream>>>(X, Wcat3, Z, ZW3);
    fill_kernel<<<N_NODES / 256, 256, 0, stream>>>(MK3, N_NODES, 0.0f);
    fill_kernel<<<N_NODES / 256, 256, 0, stream>>>(EM3, N_NODES, NEGINF);
    edge1<<<E_EDGES / 256, 256, 0, stream>>>(Z, ei, eattr, alpha + 8, MK3, EM3);
    combine1<<<N_NODES / 256, 256, 0, stream>>>(MK3, EM3, Z, deg, b3, E3v);

    // ---- fused kNN interpolate of (x + e3) ----
    knn_out_k<<<NH_PTS / 256, 256, 0, stream>>>(pos_high, pos, x, E3v, (float*)d_out);
}